// NoncommutativeKATRepresentation_38585986187669
// MI455X (gfx1250) — compile-verified
//
#include <hip/hip_runtime.h>

typedef __attribute__((ext_vector_type(2))) float v2f;
typedef __attribute__((ext_vector_type(8))) float v8f;

#define D_DIM 100
#define CRITF 15.0f
#define GAMMAF 0.2f
#define DELTAF 0.03f
#define QPT 7            // max q-stripes per thread (q = k16 + 16*j)
#define ROWS_PER_BLK 16
#define PI_F 3.14159265358979323846f

#if __has_builtin(__builtin_amdgcn_global_load_async_to_lds_b128) && \
    __has_builtin(__builtin_amdgcn_s_wait_asynccnt)
#define USE_ASYNC_LDS 1
typedef int v4i __attribute__((vector_size(16)));
typedef __attribute__((address_space(1))) v4i glb_v4i;   // global int4
typedef __attribute__((address_space(3))) v4i lds_v4i;   // LDS int4
#endif

__global__ __launch_bounds__(256)
void kat_noncomm_kernel(const float* __restrict__ x,
                        const float* __restrict__ W,
                        float* __restrict__ out)
{
    __shared__ __align__(16) float xs[ROWS_PER_BLK * D_DIM];  // 6.4 KB: x tile
    __shared__ __align__(16) float wl[D_DIM * D_DIM];         // 40 KB: raw W
    __shared__ float part[ROWS_PER_BLK][17];                  // tanh partials
    __shared__ float xmean[ROWS_PER_BLK];

    const int t  = threadIdx.x;
    const int b0 = blockIdx.x * ROWS_PER_BLK;

    // ---- stage x tile (400 x b128) and raw W (2500 x b128) into LDS ----
#ifdef USE_ASYNC_LDS
    for (int i = t; i < (ROWS_PER_BLK * D_DIM) / 4; i += 256)
        __builtin_amdgcn_global_load_async_to_lds_b128(
            (glb_v4i*)(x + (size_t)b0 * D_DIM + 4 * i),
            (lds_v4i*)(&xs[4 * i]), 0, 0);
    for (int i = t; i < (D_DIM * D_DIM) / 4; i += 256)
        __builtin_amdgcn_global_load_async_to_lds_b128(
            (glb_v4i*)(W + 4 * i),
            (lds_v4i*)(&wl[4 * i]), 0, 0);
#else
    for (int i = t; i < ROWS_PER_BLK * D_DIM; i += 256)
        xs[i] = x[(size_t)b0 * D_DIM + i];
    for (int i = t; i < D_DIM * D_DIM; i += 256)
        wl[i] = W[i];
#endif

    // ---- thread mapping: row bl (0..15), stripe k16 (0..15) ----
    const int l   = t & 31;
    const int w   = t >> 5;
    const int bl  = ((l >> 4) << 3) | (l & 7);
    const int k16 = (w << 1) | ((l >> 3) & 1);

    // ---- per-thread c_q = coeff_q * exp(-delta*k^2) (factored out of m-sum),
    //      overlapped with the async staging ----
    float acc[QPT];
    float pk[QPT];
    float cq[QPT];
    #pragma unroll
    for (int j = 0; j < QPT; ++j) {
        int   q  = k16 + 16 * j;
        float kq = (float)(q + 1);
        acc[j] = 0.0f;
        pk[j]  = PI_F * kq;                       // pi * (q+1)
        float c;
        if (q < 15) {
            c = 1.0f / kq;
        } else {
            float fq = (float)q;
            float S  = 1.0f + GAMMAF * __logf(fq / CRITF) *
                       (1.0f - __expf(-DELTAF * (fq - CRITF)));
            c = 1.0f / (kq * S);
        }
        cq[j] = c * __expf(-DELTAF * kq * kq);
    }

#ifdef USE_ASYNC_LDS
    __builtin_amdgcn_s_wait_asynccnt(0);
#endif
    __syncthreads();

    // ---- acc[j] = sum_m sin(pi*(q+1)*x[b,m]) * W[q,m], b128 LDS reads ----
    const float* xrow = &xs[bl * D_DIM];
    for (int m = 0; m < D_DIM; m += 4) {
        float4 xv = *(const float4*)&xrow[m];
        #pragma unroll
        for (int j = 0; j < QPT; ++j) {
            int q = k16 + 16 * j;
            if (q < D_DIM) {                       // uniform per wave
                float4 wv = *(const float4*)&wl[q * D_DIM + m];
                acc[j] = fmaf(__sinf(pk[j] * xv.x), wv.x, acc[j]);
                acc[j] = fmaf(__sinf(pk[j] * xv.y), wv.y, acc[j]);
                acc[j] = fmaf(__sinf(pk[j] * xv.z), wv.z, acc[j]);
                acc[j] = fmaf(__sinf(pk[j] * xv.w), wv.w, acc[j]);
            }
        }
    }

    // ---- apply c_q, per-thread tanh partial for x_mean ----
    float partial = 0.0f;
    #pragma unroll
    for (int j = 0; j < QPT; ++j) {
        int q = k16 + 16 * j;
        acc[j] *= cq[j];                           // inner_sum[b,q]
        if (q < D_DIM) partial += tanhf(acc[j]);
    }
    part[bl][k16] = partial;
    __syncthreads();

    // ---- x_mean via V_WMMA_F32_16X16X4_F32: D = A(16x16 partials) * B(1/100)
    // 4 chained K=4 WMMAs; every wave computes it (EXEC all ones, wave 0 then
    // publishes the result to all threads via LDS xmean).
    const int M  = l & 15;
    const int kh = (l >> 4) << 1;      // lanes 0-15 -> K {0,1}; 16-31 -> K {2,3}
    v8f dreg = {};
    v2f bvec = {0.01f, 0.01f};         // 1/D: turns the sum into the mean
    #pragma unroll
    for (int c4 = 0; c4 < 4; ++c4) {
        v2f a;
        a.x = part[M][4 * c4 + kh];
        a.y = part[M][4 * c4 + kh + 1];
        dreg = __builtin_amdgcn_wmma_f32_16x16x4_f32(
            false, a, false, bvec, (short)0, dreg, false, false);
    }
    // D layout: lanes 0-15 hold x_mean[0..7] in dreg[0..7]; lanes 16-31 hold [8..15]
    if (w == 0 && l == 0) {
        #pragma unroll
        for (int r = 0; r < 8; ++r) xmean[r] = dreg[r];
    }
    if (w == 0 && l == 16) {
        #pragma unroll
        for (int r = 0; r < 8; ++r) xmean[8 + r] = dreg[r];
    }
    __syncthreads();

    // ---- out[b,q] = T_q(x_mean) * inner_sum/100 ; T_q(x)=cos(q*acos(x)), |x|<1
    float xm = xmean[bl];
    float ac = acosf(xm);
    #pragma unroll
    for (int j = 0; j < QPT; ++j) {
        int q = k16 + 16 * j;
        if (q < D_DIM) {
            float Tq = __cosf((float)q * ac);
            out[(size_t)(b0 + bl) * D_DIM + q] = Tq * acc[j] * 0.01f;
        }
    }
}

extern "C" void kernel_launch(void* const* d_in, const int* in_sizes, int n_in,
                              void* d_out, int out_size, void* d_ws, size_t ws_size,
                              hipStream_t stream) {
    const float* x = (const float*)d_in[0];        // (B, 100) f32
    const float* W = (const float*)d_in[1];        // (100, 100) f32
    // d_in[2] = theta (dead: corr == 0), d_in[3] = dimension (fixed 100)
    float* out = (float*)d_out;                    // (B, 100) f32

    int Brows = in_sizes[0] / D_DIM;               // 4096
    int grid  = Brows / ROWS_PER_BLK;              // 256 blocks x 256 threads
    kat_noncomm_kernel<<<grid, 256, 0, stream>>>(x, W, out);
}